// ConvSTFT_36773509989226
// MI455X (gfx1250) — compile-verified
//
#include <hip/hip_runtime.h>

typedef __attribute__((ext_vector_type(2))) float v2f;
typedef __attribute__((ext_vector_type(8))) float v8f;

#define DIM2     1026   // 2*513 output rows
#define WINL     1024   // K
#define HOP      256
#define TFRAMES  628
#define XLEN     160000
#define PADL     768

__device__ __forceinline__ int imin(int a, int b) { return a < b ? a : b; }
__device__ __forceinline__ int imax(int a, int b) { return a > b ? a : b; }

// Branchless guarded load of F[kk][col], F[kk+1][col]:
// clamp the x index into range, load unconditionally, select 0 for invalid.
__device__ __forceinline__ v2f load_b_guard(const float* __restrict__ xb, int col, int kk) {
    const int idx  = col * HOP + kk - PADL;
    const bool cok = (col < TFRAMES);
    const bool ok0 = cok & (idx >= 0)     & (idx < XLEN);
    const bool ok1 = cok & (idx + 1 >= 0) & (idx + 1 < XLEN);
    const int  i0  = imin(imax(idx, 0),     XLEN - 1);
    const int  i1  = imin(imax(idx + 1, 0), XLEN - 1);
    const float t0 = xb[i0];
    const float t1 = xb[i1];
    v2f r;
    r.x = ok0 ? t0 : 0.0f;
    r.y = ok1 ? t1 : 0.0f;
    return r;
}

#define WMMA4(A, B0, B1, B2, B3)                                                                    \
    acc0 = __builtin_amdgcn_wmma_f32_16x16x4_f32(false, A, false, B0, (short)0, acc0, false, false); \
    acc1 = __builtin_amdgcn_wmma_f32_16x16x4_f32(false, A, false, B1, (short)0, acc1, false, false); \
    acc2 = __builtin_amdgcn_wmma_f32_16x16x4_f32(false, A, false, B2, (short)0, acc2, false, false); \
    acc3 = __builtin_amdgcn_wmma_f32_16x16x4_f32(false, A, false, B3, (short)0, acc3, false, false)

// grid: x = col groups (64 cols each, 10), y = row groups (128 rows each, 9), z = batch (64)
// block: 256 threads = 8 waves; wave w handles row-tile (by*8 + w), 16 rows x 64 cols.
__global__ __launch_bounds__(256)
void conv_stft_wmma_f32(const float* __restrict__ x,
                        const float* __restrict__ V,
                        float* __restrict__ out) {
    const int lane = threadIdx.x & 31;
    const int wave = threadIdx.x >> 5;

    const int b       = blockIdx.z;
    const int rowtile = blockIdx.y * 8 + wave;
    const int row0    = rowtile * 16;
    if (row0 >= DIM2) return;            // wave-uniform: EXEC stays all-ones for WMMA

    const int col0 = blockIdx.x * 64;

    const int m    = lane & 15;          // row-in-tile for A, col-in-tile for B/D
    const int half = lane >> 4;          // lane half selects K pair / M offset
    const int koff = half * 2;           // K offset within 4-wide step

    // A: V[row][k], row-major, stride 1024. Clamp row for the 1026 tail tile.
    const int rowA = imin(row0 + m, DIM2 - 1);
    const float* __restrict__ aptr = V + (size_t)rowA * WINL + koff;

    const float* __restrict__ xb = x + (size_t)b * XLEN;

    v8f acc0 = {}, acc1 = {}, acc2 = {}, acc3 = {};

    const int c0 = col0 + m;             // lane's column for sub-tile j=0; +16 per j

    // Interior column groups: every k of every column maps inside x (no pad / no tail)
    const bool fast = (col0 >= 3) && (col0 + 63 <= 623);

    if (fast) {
        const float* __restrict__ bp0 = xb + (size_t)(c0 +  0) * HOP - PADL + koff;
        const float* __restrict__ bp1 = xb + (size_t)(c0 + 16) * HOP - PADL + koff;
        const float* __restrict__ bp2 = xb + (size_t)(c0 + 32) * HOP - PADL + koff;
        const float* __restrict__ bp3 = xb + (size_t)(c0 + 48) * HOP - PADL + koff;

        // Software pipeline: loads for step k+4 are issued before the WMMAs on step k.
        v2f a  = *(const v2f*)(aptr);
        v2f b0 = *(const v2f*)(bp0);
        v2f b1 = *(const v2f*)(bp1);
        v2f b2 = *(const v2f*)(bp2);
        v2f b3 = *(const v2f*)(bp3);
#pragma unroll 4
        for (int k = 0; k < WINL - 4; k += 4) {
            const v2f an  = *(const v2f*)(aptr + k + 4);
            const v2f b0n = *(const v2f*)(bp0 + k + 4);
            const v2f b1n = *(const v2f*)(bp1 + k + 4);
            const v2f b2n = *(const v2f*)(bp2 + k + 4);
            const v2f b3n = *(const v2f*)(bp3 + k + 4);
            WMMA4(a, b0, b1, b2, b3);
            a = an; b0 = b0n; b1 = b1n; b2 = b2n; b3 = b3n;
        }
        WMMA4(a, b0, b1, b2, b3);
    } else {
#pragma unroll 4
        for (int k = 0; k < WINL; k += 4) {
            const v2f a  = *(const v2f*)(aptr + k);
            const int kk = k + koff;
            const v2f b0 = load_b_guard(xb, c0 +  0, kk);
            const v2f b1 = load_b_guard(xb, c0 + 16, kk);
            const v2f b2 = load_b_guard(xb, c0 + 32, kk);
            const v2f b3 = load_b_guard(xb, c0 + 48, kk);
            WMMA4(a, b0, b1, b2, b3);
        }
    }

    // D layout: VGPR v holds row M = v + 8*half, column N = lane&15.
    const int rbase = row0 + half * 8;
    float* __restrict__ outb = out + (size_t)b * DIM2 * TFRAMES;

#pragma unroll
    for (int j = 0; j < 4; ++j) {
        const int col = c0 + 16 * j;
        if (col < TFRAMES) {
            const v8f acc = (j == 0) ? acc0 : (j == 1) ? acc1 : (j == 2) ? acc2 : acc3;
#pragma unroll
            for (int v = 0; v < 8; ++v) {
                const int row = rbase + v;
                if (row < DIM2) outb[(size_t)row * TFRAMES + col] = acc[v];
            }
        }
    }
}

extern "C" void kernel_launch(void* const* d_in, const int* in_sizes, int n_in,
                              void* d_out, int out_size, void* d_ws, size_t ws_size,
                              hipStream_t stream) {
    const float* x = (const float*)d_in[0];   // (64, 160000) f32
    const float* V = (const float*)d_in[1];   // (1026, 1, 1024) f32
    float* out = (float*)d_out;               // (64, 1026, 628) f32 (viewed as (64,2,513,628))

    dim3 grid(10, 9, 64);   // 10 col-groups of 64, 9 row-groups of 128, 64 batches
    dim3 block(256);
    hipLaunchKernelGGL(conv_stft_wmma_f32, grid, block, 0, stream, x, V, out);
}